// CenterLoss_53102975648037
// MI455X (gfx1250) — compile-verified
//
#include <hip/hip_runtime.h>
#include <math.h>
#include <stdint.h>

typedef float v2f __attribute__((ext_vector_type(2)));
typedef float v8f __attribute__((ext_vector_type(8)));

constexpr int Bn   = 4096;   // rows of x
constexpr int Cn   = 8192;   // number of centers / classes
constexpr int Dn   = 128;    // feature dim (WMMA K loop: 32 steps of K=4)
constexpr int KTOP = 10;
constexpr int LSTR = 132;    // padded LDS row stride (132 % 64 == 4 -> conflict-free)

__device__ __forceinline__ float clipd(float d) {
  return fminf(fmaxf(d, 1e-12f), 1e12f);
}

// Async global -> LDS copy of 16 bytes (CDNA5 path, tracked by ASYNCcnt).
// lds_byte: LDS byte address (low 32 bits of flat shared pointer),
// base: uniform 64-bit base (SGPR pair), goff: per-lane byte offset.
__device__ __forceinline__ void async_b128(unsigned lds_byte,
                                           const float* __restrict__ base,
                                           unsigned goff) {
  asm volatile("global_load_async_to_lds_b128 %0, %1, %2"
               :
               : "v"(lds_byte), "v"(goff), "s"(base)
               : "memory");
}
__device__ __forceinline__ void wait_async_le16() {
  asm volatile("s_wait_asynccnt 0x10" ::: "memory");
}
__device__ __forceinline__ void wait_async_0() {
  asm volatile("s_wait_asynccnt 0x0" ::: "memory");
}

__device__ __forceinline__ v8f wmma_f32(v2f a, v2f b, v8f c) {
  return __builtin_amdgcn_wmma_f32_16x16x4_f32(false, a, false, b, (short)0, c,
                                               false, false);
}

// ---------------------------------------------------------------- norms ----
__global__ __launch_bounds__(256) void norms_kernel(
    const float* __restrict__ x, const float* __restrict__ cen,
    float* __restrict__ xn, float* __restrict__ cn) {
  int i = blockIdx.x * blockDim.x + threadIdx.x;
  if (i < Cn) {
    const float4* p = (const float4*)(cen + (size_t)i * Dn);
    float s = 0.f;
#pragma unroll
    for (int k = 0; k < Dn / 4; ++k) {
      float4 v = p[k];
      s += v.x * v.x + v.y * v.y + v.z * v.z + v.w * v.w;
    }
    cn[i] = s;
  }
  if (i < Bn) {
    const float4* p = (const float4*)(x + (size_t)i * Dn);
    float s = 0.f;
#pragma unroll
    for (int k = 0; k < Dn / 4; ++k) {
      float4 v = p[k];
      s += v.x * v.x + v.y * v.y + v.z * v.z + v.w * v.w;
    }
    xn[i] = s;
  }
}

// ------------------------------------------------------------ cent2cent ----
// grid = (Cn/64, Cn/64), block = 256 threads. 64x64 output tile per block:
// 8 waves, each computes one 16-row group against two 16-col groups
// (2 accumulators, 64 v_wmma_f32_16x16x4_f32 per wave).
__global__ __launch_bounds__(256) void c2c_kernel(
    const float* __restrict__ cen, const float* __restrict__ cn,
    float* __restrict__ outc, float* __restrict__ disp) {
  __shared__ float as[64 * LSTR];
  __shared__ float bs[64 * LSTR];
  __shared__ float red[256];
  const int tid  = threadIdx.x;
  const int wave = tid >> 5, lane = tid & 31, hp = lane >> 4, ln = lane & 15;
  const int rb = blockIdx.x * 64;
  const int cb = blockIdx.y * 64;

  const unsigned as_base = (unsigned)(uintptr_t)&as[0];
  const unsigned bs_base = (unsigned)(uintptr_t)&bs[0];
  for (int f = tid; f < 64 * 32; f += 256) {
    int r = f >> 5, c4 = (f & 31) << 2;
    unsigned loff = (unsigned)((r * LSTR + c4) * 4);
    async_b128(as_base + loff, cen, (unsigned)(((rb + r) * Dn + c4) * 4));
    async_b128(bs_base + loff, cen, (unsigned)(((cb + r) * Dn + c4) * 4));
  }
  wait_async_0();
  __syncthreads();

  // f32 WMMA A/B layout: lanes 0-15 hold K={k0,k0+1}, lanes 16-31 K={k0+2,k0+3}
  const int koff = hp << 1;
  const int mg = wave & 3;         // row group 0..3
  const int ng = (wave >> 2) * 2;  // col groups ng, ng+1
  const float* ap = &as[(mg * 16 + ln) * LSTR + koff];
  const float* b0 = &bs[(ng * 16 + ln) * LSTR + koff];
  const float* b1 = b0 + 16 * LSTR;
  v8f acc0 = {}, acc1 = {};
#pragma unroll
  for (int k0 = 0; k0 < Dn; k0 += 4) {
    v2f a   = *(const v2f*)(ap + k0);
    v2f bb0 = *(const v2f*)(b0 + k0);
    v2f bb1 = *(const v2f*)(b1 + k0);
    acc0 = wmma_f32(a, bb0, acc0);
    acc1 = wmma_f32(a, bb1, acc1);
  }

  const int gcol0 = cb + ng * 16 + ln;
  const int gcol1 = gcol0 + 16;
  const float cnj0 = cn[gcol0];
  const float cnj1 = cn[gcol1];
  float tsum = 0.f;
#pragma unroll
  for (int k = 0; k < 8; ++k) {
    int grow = rb + mg * 16 + k + hp * 8;  // C/D layout: vgpr k -> row k + 8*hp
    float cni = cn[grow];
    float d0 = clipd(cni + cnj0 - 2.0f * acc0[k]);
    float d1 = clipd(cni + cnj1 - 2.0f * acc1[k]);
    outc[(size_t)grow * Cn + gcol0] = d0;
    outc[(size_t)grow * Cn + gcol1] = d1;
    if (grow > gcol0) tsum += d0;  // strict lower triangle
    if (grow > gcol1) tsum += d1;
  }
  red[tid] = tsum;
  __syncthreads();
  if (tid == 0) {
    float s = 0.f;
    for (int t = 0; t < 256; ++t) s += red[t];
    disp[(size_t)blockIdx.y * gridDim.x + blockIdx.x] = s;
  }
}

// ------------------------------------------------ x->centers loss kernel ----
// grid = Bn/16 blocks, 128 threads; 128 chunks of 64 centers, double-buffered
// via async global->LDS loads (issue chunk n+1, s_wait_asynccnt 16 for chunk n).
__global__ __launch_bounds__(128) void xloss_kernel(
    const float* __restrict__ x, const int* __restrict__ labels,
    const float* __restrict__ cen, const float* __restrict__ xn,
    const float* __restrict__ cn, float* __restrict__ pullp,
    float* __restrict__ lossp) {
  __shared__ float xs[16 * LSTR];
  __shared__ float bs[2][64 * LSTR];
  __shared__ float tile[4 * 16 * 17];  // per-wave 16x16 dist tile, padded
  __shared__ float tkl[16 * 4 * KTOP];
  __shared__ float pulls[16];
  __shared__ float xnl[16];
  __shared__ int   labs[16];
  __shared__ float lsum[16], psum[16];

  const int tid  = threadIdx.x;
  const int wave = tid >> 5, lane = tid & 31, hp = lane >> 4, ln = lane & 15;
  const int rb = blockIdx.x * 16;

  const unsigned bsb0 = (unsigned)(uintptr_t)&bs[0][0];
  const unsigned bsb1 = (unsigned)(uintptr_t)&bs[1][0];

  auto issue_chunk = [&](int ch, int buf) {
    const unsigned base = buf ? bsb1 : bsb0;
    const int cbase = ch * 64;
    for (int f = tid; f < 64 * 32; f += 128) {  // 16 x b128 per thread
      int r = f >> 5, c4 = (f & 31) << 2;
      async_b128(base + (unsigned)((r * LSTR + c4) * 4), cen,
                 (unsigned)(((cbase + r) * Dn + c4) * 4));
    }
  };

  issue_chunk(0, 0);  // prologue: chunk 0 -> buffer 0

  for (int f = tid; f < 16 * 32; f += 128) {
    int r = f >> 5, c4 = (f & 31) << 2;
    *(float4*)&xs[r * LSTR + c4] = *(const float4*)(x + (size_t)(rb + r) * Dn + c4);
  }
  if (tid < 16) {
    labs[tid]  = labels[rb + tid];
    xnl[tid]   = xn[rb + tid];
    pulls[tid] = 0.f;
  }

  float tkv[KTOP];  // per-(row,wave) 10 smallest distances (label excluded)
#pragma unroll
  for (int t = 0; t < KTOP; ++t) tkv[t] = 3.0e38f;

  const int koff = hp << 1;
  const float* ap = &xs[ln * LSTR + koff];

  constexpr int NCH = Cn / 64;
  for (int ch = 0; ch < NCH; ++ch) {
    const int cur = ch & 1;
    __syncthreads();  // prev chunk's bs/tile reads done; overwrite of bs[1-cur] OK
    if (ch + 1 < NCH) {
      issue_chunk(ch + 1, cur ^ 1);
      wait_async_le16();  // in-order completion: <=16 outstanding => chunk ch done
    } else {
      wait_async_0();
    }
    __syncthreads();  // chunk ch's LDS writes visible to all waves

    const int cbase = ch * 64;
    const float* bp = &bs[cur][(wave * 16 + ln) * LSTR + koff];
    v8f acc = {};
#pragma unroll
    for (int k0 = 0; k0 < Dn; k0 += 4) {
      v2f a = *(const v2f*)(ap + k0);
      v2f b = *(const v2f*)(bp + k0);
      acc = wmma_f32(a, b, acc);
    }

    const int gcol = cbase + wave * 16 + ln;
    const float cnj = cn[gcol];
#pragma unroll
    for (int k = 0; k < 8; ++k) {
      int r = k + hp * 8;
      tile[wave * 272 + r * 17 + ln] = clipd(xnl[r] + cnj - 2.0f * acc[k]);
    }
    __syncthreads();

    if (lane < 16) {  // lane == row it owns (within this wave's 16 columns)
      const int row = lane;
      const int mylab = labs[row];
#pragma unroll
      for (int n = 0; n < 16; ++n) {
        int col = cbase + wave * 16 + n;
        float d = tile[wave * 272 + row * 17 + n];
        if (col == mylab) {
          pulls[row] = d;  // dist_pull (exactly one hit per row)
        } else {
          float mv = tkv[0];
          int   mp = 0;
#pragma unroll
          for (int t = 1; t < KTOP; ++t)
            if (tkv[t] > mv) { mv = tkv[t]; mp = t; }
          if (d < mv) tkv[mp] = d;  // keep 10 smallest distances
        }
      }
    }
  }

  if (lane < 16) {
#pragma unroll
    for (int t = 0; t < KTOP; ++t)
      tkl[lane * (4 * KTOP) + wave * KTOP + t] = tkv[t];
  }
  __syncthreads();

  if (tid < 16) {
    const int row = tid;
    float* cand = &tkl[row * (4 * KTOP)];
    float sel[KTOP];
    for (int s = 0; s < KTOP; ++s) {  // 10-pass selection of global 10 smallest
      float mv = 3.0e38f;
      int   mp = 0;
      for (int t = 0; t < 4 * KTOP; ++t) {
        float v = cand[t];
        if (v < mv) { mv = v; mp = t; }
      }
      sel[s] = mv;
      cand[mp] = 3.0e38f;
    }
    float dpull = pulls[row];
    float pos = -dpull;
    float m = fmaxf(pos, -sel[0]);  // sel[0] = smallest dist = largest neg logit
    float ssum = expf(pos - m);
    for (int t = 0; t < KTOP; ++t) ssum += expf(-sel[t] - m);
    lsum[row] = m + logf(ssum) - pos;  // -log_softmax(cat)[0]
    psum[row] = dpull;
  }
  __syncthreads();
  if (tid == 0) {
    float ls = 0.f, ps = 0.f;
    for (int r = 0; r < 16; ++r) { ls += lsum[r]; ps += psum[r]; }
    lossp[blockIdx.x] = ls;
    pullp[blockIdx.x] = ps;
  }
}

// --------------------------------------------------------- final reduce ----
__global__ __launch_bounds__(256) void reduce_kernel(
    const float* __restrict__ disp, int ndisp, const float* __restrict__ pullp,
    const float* __restrict__ lossp, int nblk, float* __restrict__ out) {
  __shared__ float red[256];
  int tid = threadIdx.x;
  float s = 0.f;
  for (int i = tid; i < ndisp; i += 256) s += disp[i];
  red[tid] = s;
  __syncthreads();
  if (tid == 0) {
    float tot = 0.f;
    for (int t = 0; t < 256; ++t) tot += red[t];
    out[1] = -tot / 33550336.0f;  // n_pairs = 8192*8191/2
    float ls = 0.f, ps = 0.f;
    for (int b = 0; b < nblk; ++b) { ls += lossp[b]; ps += pullp[b]; }
    out[0] = ls / (float)Bn;                            // loss
    out[2 + (size_t)Cn * (size_t)Cn] = ps / (float)Bn;  // loss_pull
  }
}

// -------------------------------------------------------------- launcher ----
extern "C" void kernel_launch(void* const* d_in, const int* in_sizes, int n_in,
                              void* d_out, int out_size, void* d_ws,
                              size_t ws_size, hipStream_t stream) {
  (void)in_sizes; (void)n_in; (void)out_size; (void)ws_size;
  const float* x      = (const float*)d_in[0];
  const int*   labels = (const int*)d_in[1];
  const float* cen    = (const float*)d_in[2];
  float* out = (float*)d_out;
  float* ws  = (float*)d_ws;

  // workspace (floats): xn[Bn] | cn[Cn] | disp[16384] | pullp[256] | lossp[256]
  float* xn = ws;
  float* cn = ws + Bn;
  float* disp = ws + Bn + Cn;
  const int ndisp = (Cn / 64) * (Cn / 64);  // 16384 tri-sum partials
  float* pullp = disp + ndisp;
  float* lossp = pullp + (Bn / 16);

  norms_kernel<<<(Cn + 255) / 256, 256, 0, stream>>>(x, cen, xn, cn);

  dim3 g2(Cn / 64, Cn / 64);
  c2c_kernel<<<g2, 256, 0, stream>>>(cen, cn, out + 2, disp);

  xloss_kernel<<<Bn / 16, 128, 0, stream>>>(x, labels, cen, xn, cn, pullp, lossp);

  reduce_kernel<<<1, 256, 0, stream>>>(disp, ndisp, pullp, lossp, Bn / 16, out);
}